// CRF_26603027432237
// MI455X (gfx1250) — compile-verified
//
#include <hip/hip_runtime.h>
#include <math.h>
#include <stdint.h>

// Problem constants (from reference)
#define KK     4
#define NTS    6
#define TSTART 4
#define TSTOP  5

// Tiling
#define BS      256  // block size (8 wave32s)
#define CSTRIDE 40   // floats per chunk record: 36 matrix + logscale + gold (+pad)
#define GRP     8    // chunk operators folded per Phase-B1 thread

// CDNA5 async global->LDS path (probe via __has_builtin; fallback compiles clean)
#if defined(__HIP_DEVICE_COMPILE__) && defined(__gfx1250__) && \
    __has_builtin(__builtin_amdgcn_global_load_async_to_lds_b64) && \
    __has_builtin(__builtin_amdgcn_global_load_async_to_lds_b32) && \
    __has_builtin(__builtin_amdgcn_s_wait_asynccnt)
#define USE_ASYNC_LDS 1
#else
#define USE_ASYNC_LDS 0
#endif

#define AS1 __attribute__((address_space(1)))
#define AS3 __attribute__((address_space(3)))
typedef int v2i __attribute__((vector_size(8)));   // <2 x i32> for b64 async ops

// -------------------------------------------------------------------------
// Phase A: one thread per (row, chunk). Builds the chunk's 6x6 linear-domain
// transfer operator P (with accumulated log-scale ls) and partial gold score.
// 3-slot rotating LDS stage, 2 steps of async lookahead (3xb64 + 2xb32/step).
// -------------------------------------------------------------------------
__global__ __launch_bounds__(BS) void crf_chunk_kernel(
    const float* __restrict__ feats, const float* __restrict__ bias,
    const int*   __restrict__ tags,  const float* __restrict__ trans,
    const float* __restrict__ w_shift, const float* __restrict__ bias_no,
    const float* __restrict__ bias_with, const float* __restrict__ w_with,
    const float* __restrict__ w_no, const float* __restrict__ mult,
    float* __restrict__ ws, int Bv, int T, int NCv, int CTv)
{
    __shared__ float Msh[KK * KK];     // Mmod for dynamic (t1,t0) indexing
    __shared__ float Tsh[NTS * NTS];   // transitions for dynamic indexing
#if USE_ASYNC_LDS
    // 10 floats/lane/slot: feats[0..5], bias[6], tag-bits[7], 2 pad.
    // 40B stride: 8B-aligned for b64; 32 lanes hit 32 distinct banks.
    __shared__ float fstage[3][BS][10];
#endif

    const int tid = threadIdx.x;
    const int gid = blockIdx.x * BS + tid;

    // ---- uniform parameter prep (scalar loads; lives in SGPRs) ----
    // Mmod = softmax(multiplier, axis=0); then diag = -1
    float Ms[KK][KK];
#pragma unroll
    for (int j = 0; j < KK; ++j) {           // column j, normalize over rows i
        float mx = -3.4e38f;
#pragma unroll
        for (int i = 0; i < KK; ++i) mx = fmaxf(mx, mult[i * KK + j]);
        float e[KK]; float ssum = 0.f;
#pragma unroll
        for (int i = 0; i < KK; ++i) { e[i] = __expf(mult[i * KK + j] - mx); ssum += e[i]; }
        const float inv = 1.f / ssum;
#pragma unroll
        for (int i = 0; i < KK; ++i) Ms[i][j] = e[i] * inv;
    }
#pragma unroll
    for (int d = 0; d < KK; ++d) Ms[d][d] = -1.f;

    // exp(transitions): -10000 rows/cols underflow to exact 0 (acts as -inf)
    float eT[NTS][NTS];
#pragma unroll
    for (int j = 0; j < NTS; ++j)
#pragma unroll
        for (int i = 0; i < NTS; ++i) eT[j][i] = __expf(trans[j * NTS + i]);

    float wsh[KK], wwi[KK], wno[KK];
#pragma unroll
    for (int k = 0; k < KK; ++k) { wsh[k] = w_shift[k]; wwi[k] = w_with[k]; wno[k] = w_no[k]; }
    const float bw = bias_with[0], bn = bias_no[0];

    if (tid == 0) {
        for (int j = 0; j < KK; ++j)
            for (int i = 0; i < KK; ++i) Msh[j * KK + i] = Ms[j][i];
        for (int x = 0; x < NTS * NTS; ++x) Tsh[x] = trans[x];
    }
    __syncthreads();

    if (gid >= Bv * NCv) return;

    const int b  = gid / NCv;
    const int c  = gid - b * NCv;
    const int s0 = c * CTv;
    const int rem = T - s0;
    const int steps = (CTv < rem) ? CTv : rem;

    const size_t rowoff = (size_t)b * (size_t)T + (size_t)s0;
    const float* fp = feats + rowoff * NTS;
    const float* bp = bias + rowoff;
    const int*   tp = tags + rowoff;

    // chunk operator = identity
    float P[NTS][NTS];
#pragma unroll
    for (int j = 0; j < NTS; ++j)
#pragma unroll
        for (int i = 0; i < NTS; ++i) P[j][i] = (i == j) ? 1.f : 0.f;
    float ls = 0.f, gold = 0.f;
    int t0 = (s0 == 0) ? TSTART : tp[-1];

#if USE_ASYNC_LDS
    // One step's data = 5 async ops into this lane's slot.
#define ISSUE_STEP(q_, i_) do {                                               \
        const float* _f = fp + (size_t)(i_) * NTS;                            \
        char* _d = (char*)(&fstage[(q_)][tid][0]);                            \
        __builtin_amdgcn_global_load_async_to_lds_b64(                        \
            (AS1 v2i*)(uintptr_t)(_f + 0), (AS3 v2i*)(uintptr_t)(_d + 0), 0, 0); \
        __builtin_amdgcn_global_load_async_to_lds_b64(                        \
            (AS1 v2i*)(uintptr_t)(_f + 2), (AS3 v2i*)(uintptr_t)(_d + 8), 0, 0); \
        __builtin_amdgcn_global_load_async_to_lds_b64(                        \
            (AS1 v2i*)(uintptr_t)(_f + 4), (AS3 v2i*)(uintptr_t)(_d + 16), 0, 0); \
        __builtin_amdgcn_global_load_async_to_lds_b32(                        \
            (AS1 int*)(uintptr_t)(bp + (i_)), (AS3 int*)(uintptr_t)(_d + 24), 0, 0); \
        __builtin_amdgcn_global_load_async_to_lds_b32(                        \
            (AS1 int*)(uintptr_t)(tp + (i_)), (AS3 int*)(uintptr_t)(_d + 28), 0, 0); \
    } while (0)
    ISSUE_STEP(0, 0);
    if (steps > 1) ISSUE_STEP(1, 1);
#endif

    for (int i = 0; i < steps; ++i) {
        float f6[NTS];
        float bv;
        int   t1;
#if USE_ASYNC_LDS
        if (i + 2 < steps)      { ISSUE_STEP((i + 2) % 3, i + 2);
                                  __builtin_amdgcn_s_wait_asynccnt(10); }
        else if (i + 1 < steps) { __builtin_amdgcn_s_wait_asynccnt(5); }
        else                    { __builtin_amdgcn_s_wait_asynccnt(0); }
        {
            const int sl = i % 3;
#pragma unroll
            for (int j = 0; j < NTS; ++j) f6[j] = fstage[sl][tid][j];
            bv = fstage[sl][tid][6];
            t1 = __float_as_int(fstage[sl][tid][7]);
        }
#else
        const float* src = fp + (size_t)i * NTS;
#pragma unroll
        for (int j = 0; j < NTS; ++j) f6[j] = src[j];
        if (i + 8 < steps) __builtin_prefetch(fp + (size_t)(i + 8) * NTS, 0, 1);
        bv = bp[i];
        t1 = tp[i];
#endif

        // gates[b,t,k]
        const bool  wsel = bv > 0.5f;
        const float bb = wsel ? bw : bn;
        float g[KK];
#pragma unroll
        for (int k = 0; k < KK; ++k) {
            const float x  = bv * wsh[k] + bb;
            const float e2 = __expf(2.f * x);          // tanh via one exp
            const float th = (e2 - 1.f) / (e2 + 1.f);
            g[k] = (wsel ? wwi[k] : wno[k]) * th;
        }

        float ef[NTS];
#pragma unroll
        for (int j = 0; j < NTS; ++j) ef[j] = __expf(f6[j]);

        float eg[KK][KK];
#pragma unroll
        for (int j = 0; j < KK; ++j)
#pragma unroll
            for (int i2 = 0; i2 < KK; ++i2) eg[j][i2] = __expf(g[i2] * Ms[j][i2]);

        // P <- diag(ef) * E_t * P   (linear-domain step operator compose)
        float Pn[NTS][NTS];
#pragma unroll
        for (int j = 0; j < NTS; ++j) {
            float E[NTS];
#pragma unroll
            for (int k2 = 0; k2 < NTS; ++k2)
                E[k2] = eT[j][k2] * ((j < KK && k2 < KK) ? eg[j][k2] : 1.f);
#pragma unroll
            for (int col = 0; col < NTS; ++col) {
                float acc = 0.f;
#pragma unroll
                for (int k2 = 0; k2 < NTS; ++k2) acc = fmaf(E[k2], P[k2][col], acc);
                Pn[j][col] = ef[j] * acc;
            }
        }
#pragma unroll
        for (int j = 0; j < NTS; ++j)
#pragma unroll
            for (int col = 0; col < NTS; ++col) P[j][col] = Pn[j][col];

        // rescale every 4 steps (random-walk drift stays far from f32 limits)
        if ((i & 3) == 3) {
            float m = 0.f;
#pragma unroll
            for (int j = 0; j < NTS; ++j)
#pragma unroll
                for (int col = 0; col < NTS; ++col) m = fmaxf(m, P[j][col]);
            if (m > 0.f) {
                const float inv = 1.f / m;
#pragma unroll
                for (int j = 0; j < NTS; ++j)
#pragma unroll
                    for (int col = 0; col < NTS; ++col) P[j][col] *= inv;
                ls += __logf(m);
            }
        }

        // gold path partial: base + extra + emit
        const float base = Tsh[t1 * NTS + t0];
        float extra = 0.f;
        if (t0 < KK && t1 < KK) {
            const float gt0 = (t0 == 0) ? g[0] : (t0 == 1) ? g[1]
                              : (t0 == 2) ? g[2] : g[3];
            extra = gt0 * Msh[t1 * KK + t0];
        }
        const float emit = (t1 == 0) ? f6[0] : (t1 == 1) ? f6[1]
                         : (t1 == 2) ? f6[2] : (t1 == 3) ? f6[3]
                         : (t1 == 4) ? f6[4] : f6[5];
        gold += base + extra + emit;
        t0 = t1;
    }

    // final rescale so stored entries are well-conditioned
    {
        float m = 0.f;
#pragma unroll
        for (int j = 0; j < NTS; ++j)
#pragma unroll
            for (int col = 0; col < NTS; ++col) m = fmaxf(m, P[j][col]);
        if (m > 0.f) {
            const float inv = 1.f / m;
#pragma unroll
            for (int j = 0; j < NTS; ++j)
#pragma unroll
                for (int col = 0; col < NTS; ++col) P[j][col] *= inv;
            ls += __logf(m);
        }
    }

    float* o = ws + ((size_t)b * NCv + c) * CSTRIDE;
#pragma unroll
    for (int j = 0; j < NTS; ++j)
#pragma unroll
        for (int i = 0; i < NTS; ++i) o[j * NTS + i] = P[j][i];
    o[36] = ls;
    o[37] = gold;
}

// -------------------------------------------------------------------------
// Phase B1 (tree level): one thread per (row, group of GRP chunks); folds
// GRP chunk operators into one via 6x6 matrix products (time order, rescaled).
// Gives B*NC/GRP threads reading the big level-1 table at good occupancy.
// -------------------------------------------------------------------------
__global__ __launch_bounds__(256) void crf_partial_combine_kernel(
    const float* __restrict__ tab1, float* __restrict__ tab2,
    int Bv, int NCv, int NGv)
{
    const int idx = blockIdx.x * blockDim.x + threadIdx.x;
    if (idx >= Bv * NGv) return;
    const int b = idx / NGv;
    const int g = idx - b * NGv;

    const int c0 = g * GRP;
    const int c1 = (c0 + GRP < NCv) ? (c0 + GRP) : NCv;

    float A[NTS][NTS];
#pragma unroll
    for (int j = 0; j < NTS; ++j)
#pragma unroll
        for (int i = 0; i < NTS; ++i) A[j][i] = (i == j) ? 1.f : 0.f;
    float ls = 0.f, gold = 0.f;

    for (int c = c0; c < c1; ++c) {
        const float* m = tab1 + ((size_t)b * NCv + c) * CSTRIDE;
        float Q[NTS][NTS];
#pragma unroll
        for (int j = 0; j < NTS; ++j)
#pragma unroll
            for (int col = 0; col < NTS; ++col) {
                float acc = 0.f;
#pragma unroll
                for (int k = 0; k < NTS; ++k) acc = fmaf(m[j * NTS + k], A[k][col], acc);
                Q[j][col] = acc;
            }
        ls += m[36];
        gold += m[37];
        float mx = 0.f;
#pragma unroll
        for (int j = 0; j < NTS; ++j)
#pragma unroll
            for (int col = 0; col < NTS; ++col) mx = fmaxf(mx, Q[j][col]);
        if (mx > 0.f) {
            const float inv = 1.f / mx;
#pragma unroll
            for (int j = 0; j < NTS; ++j)
#pragma unroll
                for (int col = 0; col < NTS; ++col) A[j][col] = Q[j][col] * inv;
            ls += __logf(mx);
        } else {
#pragma unroll
            for (int j = 0; j < NTS; ++j)
#pragma unroll
                for (int col = 0; col < NTS; ++col) A[j][col] = Q[j][col];
        }
    }

    float* o = tab2 + ((size_t)b * NGv + g) * CSTRIDE;
#pragma unroll
    for (int j = 0; j < NTS; ++j)
#pragma unroll
        for (int i = 0; i < NTS; ++i) o[j * NTS + i] = A[j][i];
    o[36] = ls;
    o[37] = gold;
}

// -------------------------------------------------------------------------
// Phase B2: one thread per row; folds operators in time order into the
// init vector, applies STOP transition, subtracts gold.
// -------------------------------------------------------------------------
__global__ __launch_bounds__(256) void crf_combine_kernel(
    const float* __restrict__ tab, const float* __restrict__ trans,
    const int* __restrict__ tags, float* __restrict__ out,
    int Bv, int T, int NR)
{
    const int b = blockIdx.x * blockDim.x + threadIdx.x;
    if (b >= Bv) return;

    float p[NTS];
#pragma unroll
    for (int j = 0; j < NTS; ++j) p[j] = (j == TSTART) ? 1.f : 0.f;  // exp(init)
    float ls = 0.f, gold = 0.f;

    const float* basep = tab + (size_t)b * NR * CSTRIDE;
    for (int c = 0; c < NR; ++c) {
        const float* m = basep + (size_t)c * CSTRIDE;
        float q[NTS];
#pragma unroll
        for (int j = 0; j < NTS; ++j) {
            float acc = 0.f;
#pragma unroll
            for (int i = 0; i < NTS; ++i) acc = fmaf(m[j * NTS + i], p[i], acc);
            q[j] = acc;
        }
        ls += m[36];
        gold += m[37];
        float mx = 0.f;
#pragma unroll
        for (int j = 0; j < NTS; ++j) mx = fmaxf(mx, q[j]);
        if (mx > 0.f) {
            const float inv = 1.f / mx;
#pragma unroll
            for (int j = 0; j < NTS; ++j) p[j] = q[j] * inv;
            ls += __logf(mx);
        } else {
#pragma unroll
            for (int j = 0; j < NTS; ++j) p[j] = q[j];
        }
    }

    float s = 0.f;
#pragma unroll
    for (int j = 0; j < NTS; ++j) s += p[j] * __expf(trans[TSTOP * NTS + j]);
    const float fwd = __logf(s) + ls;

    gold += trans[TSTOP * NTS + tags[(size_t)b * T + (T - 1)]];
    out[b] = fwd - gold;
}

// -------------------------------------------------------------------------
extern "C" void kernel_launch(void* const* d_in, const int* in_sizes, int n_in,
                              void* d_out, int out_size, void* d_ws, size_t ws_size,
                              hipStream_t stream)
{
    const float* feats = (const float*)d_in[0];
    const float* bias  = (const float*)d_in[1];
    const int*   tags  = (const int*)  d_in[2];
    const float* trans = (const float*)d_in[3];
    const float* wshf  = (const float*)d_in[4];
    const float* bno   = (const float*)d_in[5];
    const float* bwi   = (const float*)d_in[6];
    const float* wwo   = (const float*)d_in[7];
    const float* wno   = (const float*)d_in[8];
    const float* mult  = (const float*)d_in[9];
    float* out = (float*)d_out;
    float* ws  = (float*)d_ws;

    const int B = out_size;                              // output is (B,)
    const int T = (B > 0) ? (int)(in_sizes[1] / B) : 0;  // bias is (B,T)

    // Pick the smallest chunk size whose chunk-operator table fits in d_ws.
    // CT=32 -> 4096 wave32s of Phase-A parallelism for B=2048,T=2048 (21MB ws).
    int CTv = 32, NC = (T + CTv - 1) / CTv;
    while ((size_t)B * NC * CSTRIDE * sizeof(float) > ws_size && CTv < T) {
        CTv *= 2;
        NC = (T + CTv - 1) / CTv;
    }

    const int total = B * NC;
    dim3 gridA((unsigned)((total + BS - 1) / BS)), blockA(BS);
    hipLaunchKernelGGL(crf_chunk_kernel, gridA, blockA, 0, stream,
                       feats, bias, tags, trans, wshf, bno, bwi, wwo, wno, mult,
                       ws, B, T, NC, CTv);

    // Two-level combine when scratch allows: level-1 table is read by
    // B*ceil(NC/GRP) threads instead of just B threads.
    const int NG = (NC + GRP - 1) / GRP;
    const size_t tab1_bytes = (size_t)B * NC * CSTRIDE * sizeof(float);
    const size_t tab2_bytes = (size_t)B * NG * CSTRIDE * sizeof(float);
    const bool two_level = (NC > GRP) && (tab1_bytes + tab2_bytes <= ws_size);

    if (two_level) {
        float* ws2 = ws + (size_t)B * NC * CSTRIDE;
        dim3 gridP((unsigned)((B * NG + 255) / 256)), blockP(256);
        hipLaunchKernelGGL(crf_partial_combine_kernel, gridP, blockP, 0, stream,
                           ws, ws2, B, NC, NG);
        dim3 gridB((unsigned)((B + 255) / 256)), blockB(256);
        hipLaunchKernelGGL(crf_combine_kernel, gridB, blockB, 0, stream,
                           ws2, trans, tags, out, B, T, NG);
    } else {
        dim3 gridB((unsigned)((B + 255) / 256)), blockB(256);
        hipLaunchKernelGGL(crf_combine_kernel, gridB, blockB, 0, stream,
                           ws, trans, tags, out, B, T, NC);
    }
}